// ModuleSoftsplat_48112223649935
// MI455X (gfx1250) — compile-verified
//
#include <hip/hip_runtime.h>

// Fixed problem dims from the reference:
// tenInput (4,64,256,448) f32, tenFlow (4,2,256,448), tenMetric (4,1,256,448)
#define NB 4
#define CH 64
#define HH 256
#define WW 448
#define HW (HH * WW)            // 114688
#define CHW (CH * HW)           // 7340032
#define NPIX (NB * HW)          // 458752  (== 1792 * 256 exactly)
#define OUT_ELEMS ((size_t)NB * CHW)   // 29360128 floats

// Native no-return f32 global atomic add at device scope, with a
// compile-time byte offset folded into the signed-24-bit VMEM offset field
// (channel stride 458752B * 7 = 3.2MB < 8.38MB limit). No "memory" clobber:
// nothing in the splat kernel reads the atomic targets, so the scheduler is
// free to batch the coalesced input loads ahead of the atomic burst.
template <int BYTEOFS>
__device__ __forceinline__ void gatomic_fadd_ofs(float* base, float v) {
  asm volatile("global_atomic_add_f32 %0, %1, off offset:%c2 scope:SCOPE_DEV"
               :
               : "v"((unsigned long long)base), "v"(v), "i"(BYTEOFS));
}

// ---------------------------------------------------------------------------
// Kernel 1: zero the 64-channel output accumulator (d_out) and the
// splatted-metric accumulator (d_ws). Must run every call (buffers poisoned).
// Grid is sized exactly OUT_ELEMS/4 threads.
__global__ __launch_bounds__(256) void zero_kernel(float4* __restrict__ out4,
                                                   float4* __restrict__ nrm4) {
  unsigned i = blockIdx.x * 256u + threadIdx.x;
  float4 z;
  z.x = z.y = z.z = z.w = 0.f;
  out4[i] = z;                       // exact coverage, no bounds check needed
  if (i < (unsigned)(NPIX / 4)) nrm4[i] = z;
}

// ---------------------------------------------------------------------------
// Kernel 2: forward splat. One thread per source pixel.
#if __has_builtin(__builtin_amdgcn_global_load_async_to_lds_b32)
#define USE_ASYNC_LDS 1
#endif

// One channel within an 8-channel group: 1 coalesced load, 4 offset-immediate
// atomics (one per bilinear corner).
#define SPLAT_CH(J)                                         \
  do {                                                      \
    const float v_##J = ipc[(size_t)(J) * HW] * met;        \
    gatomic_fadd_ofs<(J) * HW * 4>(a00, v_##J * w00);       \
    gatomic_fadd_ofs<(J) * HW * 4>(a10, v_##J * w10);       \
    gatomic_fadd_ofs<(J) * HW * 4>(a01, v_##J * w01);       \
    gatomic_fadd_ofs<(J) * HW * 4>(a11, v_##J * w11);       \
  } while (0)

__global__ __launch_bounds__(256) void splat_kernel(
    const float* __restrict__ inp, const float* __restrict__ flow,
    const float* __restrict__ metric, float* __restrict__ out,
    float* __restrict__ nrm) {
  const int tid = blockIdx.x * 256 + threadIdx.x;  // < NPIX (exact grid)
  const int n = tid / HW;
  const int p = tid - n * HW;
  const int y = p / WW;
  const int x = p - y * WW;

  const float* fxp = flow + (size_t)n * 2 * HW + p;
  const float* fyp = fxp + HW;
  const float* mtp = metric + tid;  // metric is (N,1,H,W) -> linear == tid

  float fx, fy, mraw;
#ifdef USE_ASYNC_LDS
  // gfx1250 async DMA: global -> LDS, tracked by ASYNCcnt. Each lane stages
  // its own 3 dwords; only per-wave asynccnt wait is needed (no barrier since
  // each lane reads back only its own slots).
  __shared__ float s_stage[3 * 256];
  typedef __attribute__((address_space(1))) int GInt;
  typedef __attribute__((address_space(3))) int LInt;
  const unsigned t = threadIdx.x;
  __builtin_amdgcn_global_load_async_to_lds_b32(
      (GInt*)(void*)fxp, (LInt*)(void*)&s_stage[t], 0, 0);
  __builtin_amdgcn_global_load_async_to_lds_b32(
      (GInt*)(void*)fyp, (LInt*)(void*)&s_stage[256 + t], 0, 0);
  __builtin_amdgcn_global_load_async_to_lds_b32(
      (GInt*)(void*)mtp, (LInt*)(void*)&s_stage[512 + t], 0, 0);
#if __has_builtin(__builtin_amdgcn_s_wait_asynccnt)
  __builtin_amdgcn_s_wait_asynccnt(0);
#else
  asm volatile("s_wait_asynccnt 0" ::: "memory");
#endif
  asm volatile("" ::: "memory");  // keep LDS reads below the wait
  fx = s_stage[t];
  fy = s_stage[256 + t];
  mraw = s_stage[512 + t];
#else
  fx = *fxp;
  fy = *fyp;
  mraw = *mtp;
#endif
  const float met = expf(mraw);

  const float xx = (float)x + fx;
  const float yyf = (float)y + fy;
  const float x0f = floorf(xx);
  const float y0f = floorf(yyf);
  const int ix0 = (int)x0f, iy0 = (int)y0f;
  const int ix1 = ix0 + 1, iy1 = iy0 + 1;
  const float ax = xx - x0f, ay = yyf - y0f;
  const float bx = 1.f - ax, by = 1.f - ay;
  float w00 = bx * by, w10 = ax * by, w01 = bx * ay, w11 = ax * ay;

  // Reference semantics: weight *= valid, index clipped. Branchless.
  const bool vx0 = (unsigned)ix0 < (unsigned)WW;
  const bool vx1 = (unsigned)ix1 < (unsigned)WW;
  const bool vy0 = (unsigned)iy0 < (unsigned)HH;
  const bool vy1 = (unsigned)iy1 < (unsigned)HH;
  w00 = (vx0 && vy0) ? w00 : 0.f;
  w10 = (vx1 && vy0) ? w10 : 0.f;
  w01 = (vx0 && vy1) ? w01 : 0.f;
  w11 = (vx1 && vy1) ? w11 : 0.f;

  const int cx0 = min(max(ix0, 0), WW - 1);
  const int cx1 = min(max(ix1, 0), WW - 1);
  const int cy0 = min(max(iy0, 0), HH - 1);
  const int cy1 = min(max(iy1, 0), HH - 1);
  const int t00 = cy0 * WW + cx0;
  const int t10 = cy0 * WW + cx1;
  const int t01 = cy1 * WW + cx0;
  const int t11 = cy1 * WW + cx1;

  float* const ob = out + (size_t)n * CHW;
  float* a00 = ob + t00;
  float* a10 = ob + t10;
  float* a01 = ob + t01;
  float* a11 = ob + t11;
  const float* ipc = inp + (size_t)n * CHW + p;

  // 8 groups of 8 channels. Per group: 1 prefetch (stride 448KB defeats any
  // HW prefetcher; speculative TH=0 drops OOB translations silently),
  // 8 coalesced loads, 32 offset-immediate atomics, then 5 pointer bumps.
#pragma unroll 1
  for (int g = 0; g < CH; g += 8) {
    __builtin_prefetch(ipc + (size_t)16 * HW, 0, 1);
    SPLAT_CH(0);
    SPLAT_CH(1);
    SPLAT_CH(2);
    SPLAT_CH(3);
    SPLAT_CH(4);
    SPLAT_CH(5);
    SPLAT_CH(6);
    SPLAT_CH(7);
    ipc += (size_t)8 * HW;
    a00 += (size_t)8 * HW;
    a10 += (size_t)8 * HW;
    a01 += (size_t)8 * HW;
    a11 += (size_t)8 * HW;
  }

  // Channel 64 of the reference's concat: the metric itself (normalizer).
  float* const nb = nrm + (size_t)n * HW;
  gatomic_fadd_ofs<0>(nb + t00, met * w00);
  gatomic_fadd_ofs<0>(nb + t10, met * w10);
  gatomic_fadd_ofs<0>(nb + t01, met * w01);
  gatomic_fadd_ofs<0>(nb + t11, met * w11);
}

// ---------------------------------------------------------------------------
// Kernel 3: normalize. One thread per pixel; one IEEE divide, 64 mul-stores.
__global__ __launch_bounds__(256) void normalize_kernel(
    float* __restrict__ out, const float* __restrict__ nrm) {
  const int tid = blockIdx.x * 256 + threadIdx.x;  // < NPIX (exact grid)
  const int n = tid / HW;
  const int p = tid - n * HW;
  const float d = nrm[tid];
  const float inv = (d == 0.f) ? 1.f : 1.f / d;
  float* const o = out + (size_t)n * CHW + p;
#pragma unroll 8
  for (int c = 0; c < CH; ++c) {
    o[(size_t)c * HW] *= inv;  // accumulator is L2-resident; coalesced
  }
}

// ---------------------------------------------------------------------------
extern "C" void kernel_launch(void* const* d_in, const int* in_sizes, int n_in,
                              void* d_out, int out_size, void* d_ws,
                              size_t ws_size, hipStream_t stream) {
  const float* inp = (const float*)d_in[0];     // (4,64,256,448)
  const float* flow = (const float*)d_in[1];    // (4,2,256,448)
  const float* metric = (const float*)d_in[2];  // (4,1,256,448)
  float* out = (float*)d_out;                   // (4,64,256,448)
  float* nrm = (float*)d_ws;                    // (4,256,448) accumulator

  (void)in_sizes; (void)n_in; (void)out_size; (void)ws_size;

  const dim3 blk(256);
  zero_kernel<<<dim3(OUT_ELEMS / 4 / 256), blk, 0, stream>>>(
      (float4*)out, (float4*)nrm);
  splat_kernel<<<dim3(NPIX / 256), blk, 0, stream>>>(inp, flow, metric, out,
                                                     nrm);
  normalize_kernel<<<dim3(NPIX / 256), blk, 0, stream>>>(out, nrm);
}